// IsoneutralMixing_63453846831738
// MI455X (gfx1250) — compile-verified
//
#include <hip/hip_runtime.h>
#include <cstdint>
#include <cstddef>

namespace {
constexpr int NX = 364, NY = 184, NZ = 40;
constexpr int NPT = NX * NY * NZ;

constexpr float RHO0     = 1024.0f;
constexpr float GRAV     = 9.81f;
constexpr float BETA_T   = 1.67e-4f;
constexpr float BETA_TS  = 1e-5f;
constexpr float GAMMAS   = 1.1e-8f;
constexpr float BETA_Sc  = 0.78e-3f;
constexpr float THETA0   = 283.0f - 273.15f;
constexpr float EPSc     = 1e-20f;
constexpr float ISO_SLOPEC = 1e-3f;
constexpr float INV_ISO_DSLOPE = 1000.0f;   // 1 / 1e-3
constexpr float K_ISO_STEEP = 50.0f;
}

typedef __attribute__((ext_vector_type(4))) float v4f;

// ---------------- device helpers -----------------------------------------

__device__ __forceinline__ size_t idx3(int i, int j, int k) {
  return ((size_t)i * NY + (size_t)j) * NZ + (size_t)k;
}

// temp/salt are (X,Y,Z,3); tau = 0 slice
__device__ __forceinline__ float ts(const float* __restrict__ a, int i, int j, int k) {
  return a[idx3(i, j, k) * 3u];
}

__device__ __forceinline__ float fast_tanh(float x) {
#if __has_builtin(__builtin_amdgcn_tanhf)
  return __builtin_amdgcn_tanhf(x);     // v_tanh_f32 on gfx1250
#else
  return tanhf(x);
#endif
}

__device__ __forceinline__ float dm_taper(float s) {
  return 0.5f * (1.0f + fast_tanh((ISO_SLOPEC - fabsf(s)) * INV_ISO_DSLOPE));
}

__device__ __forceinline__ float drhodT(float Tv, float absz) {
  return -(BETA_TS * (Tv - THETA0)
           + BETA_T * (1.0f + GAMMAS * GRAV * RHO0 * (-absz))) * RHO0;
}

// non-temporal streaming stores: outputs are write-once, never re-read;
// keep the (re-read) inputs resident in the 192MB L2 instead.
__device__ __forceinline__ void nt_store(float* p, float v) {
  __builtin_nontemporal_store(v, p);
}
__device__ __forceinline__ void nt_store4(float* p, float a, float b, float c, float d) {
  v4f v = {a, b, c, d};
  __builtin_nontemporal_store(v, (v4f*)p);   // 16B-aligned -> global_store_b128 nt
}

// Stage n floats from global g into LDS starting at byte offset lds_byte,
// using CDNA5 async global->LDS DMA (tracked by ASYNCcnt).
__device__ __forceinline__ void async_stage(uint32_t lds_byte,
                                            const float* __restrict__ g,
                                            int n, int tid, int nt) {
  for (int x = tid; x < n; x += nt) {
    uint32_t la = lds_byte + 4u * (uint32_t)x;
    const float* ga = g + x;
    asm volatile("global_load_async_to_lds_b32 %0, %1, off"
                 :: "v"(la), "v"(ga) : "memory");
  }
}

__device__ __forceinline__ void async_wait_and_sync() {
  asm volatile("s_wait_asynccnt 0" ::: "memory");
  __syncthreads();
}

// ---------------- kernel 1: K_11 and Ai_ez (east faces) -------------------

__global__ __launch_bounds__(256)
void iso_k11_aiez(const float* __restrict__ maskT, const float* __restrict__ maskU,
                  const float* __restrict__ maskW,
                  const float* __restrict__ dxu, const float* __restrict__ cost,
                  const float* __restrict__ dzt, const float* __restrict__ dzw,
                  const float* __restrict__ zt,
                  const float* __restrict__ temp, const float* __restrict__ salt,
                  const float* __restrict__ Kiso,
                  float* __restrict__ K11, float* __restrict__ Aiez)
{
  __shared__ float smem[3 * NZ + NX + NY];
  float* s_dzw  = smem;
  float* s_dzt  = smem + NZ;
  float* s_zt   = smem + 2 * NZ;
  float* s_dxu  = smem + 3 * NZ;
  float* s_cost = smem + 3 * NZ + NX;

  const int tid = threadIdx.x, nt = blockDim.x;
  const uint32_t sb = (uint32_t)(uintptr_t)smem;   // LDS byte offset of smem
  async_stage(sb,                          dzw,  NZ, tid, nt);
  async_stage(sb + 4u * NZ,                dzt,  NZ, tid, nt);
  async_stage(sb + 8u * NZ,                zt,   NZ, tid, nt);
  async_stage(sb + 12u * NZ,               dxu,  NX, tid, nt);
  async_stage(sb + 12u * NZ + 4u * NX,     cost, NY, tid, nt);
  async_wait_and_sync();

  int n = blockIdx.x * nt + tid;
  if (n >= NPT) return;
  int k = n % NZ; int r = n / NZ; int j = r % NY; int i = r / NY;
  size_t c = idx3(i, j, k);

  bool inside = (i >= 1) & (i <= NX - 3) & (j >= 2) & (j <= NY - 3);
  if (!inside) {
    nt_store(&K11[c], 0.0f);
    nt_store4(Aiez + c * 4u, 0.f, 0.f, 0.f, 0.f);
    return;
  }

  float mU  = maskU[c];
  float fxi = s_dxu[i] * s_cost[j];
  float Tc = ts(temp, i, j, k),  Te = ts(temp, i + 1, j, k);
  float Sc = ts(salt, i, j, k),  Se = ts(salt, i + 1, j, k);
  float dTdxv = mU * (Te - Tc) / fxi;
  float dSdxv = mU * (Se - Sc) / fxi;

  float diffloc = (k >= 1)
    ? 0.25f * (Kiso[idx3(i, j, k)] + Kiso[idx3(i, j, k - 1)]
             + Kiso[idx3(i + 1, j, k)] + Kiso[idx3(i + 1, j, k - 1)])
    : 0.5f * (Kiso[idx3(i, j, 0)] + Kiso[idx3(i + 1, j, 0)]);

  float sumz = 0.f;
  float ai[2][2] = {{0.f, 0.f}, {0.f, 0.f}};
  float abszk = fabsf(s_zt[k]);

  #pragma unroll
  for (int kr = 0; kr < 2; ++kr) {
    const int ki = (kr == 1) ? 0 : 1;
    if (k < ki) continue;                      // kr=0 contributes only for k>=1
    const int kz = (kr == 0) ? k - 1 : k;
    float dzwz = s_dzw[kz];
    #pragma unroll
    for (int ip = 0; ip < 2; ++ip) {
      int ii = i + ip;
      float mT   = maskT[idx3(ii, j, k)];
      float drdT = mT * drhodT(ts(temp, ii, j, k), abszk);
      float drdS = mT * (BETA_Sc * RHO0);
      float dTdzv = 0.f, dSdzv = 0.f;
      if (kz < NZ - 1) {
        float mWv = maskW[idx3(ii, j, kz)];
        dTdzv = mWv * (ts(temp, ii, j, kz + 1) - ts(temp, ii, j, kz)) / dzwz;
        dSdzv = mWv * (ts(salt, ii, j, kz + 1) - ts(salt, ii, j, kz)) / dzwz;
      }
      float drodxe = drdT * dTdxv + drdS * dSdxv;
      float drodze = drdT * dTdzv + drdS * dSdzv;
      float sxe = -drodxe / (fminf(0.f, drodze) - EPSc);
      float tap = dm_taper(sxe);
      sumz += dzwz * mU * fmaxf(K_ISO_STEEP, diffloc * tap);
      ai[ip][kr] = tap * sxe * mU;
    }
  }
  nt_store(&K11[c], sumz / (4.0f * s_dzt[k]));
  // (...,ip,kr) flat = ip*2 + kr
  nt_store4(Aiez + c * 4u, ai[0][0], ai[0][1], ai[1][0], ai[1][1]);
}

// ---------------- kernel 2: K_22 and Ai_nz (north faces) ------------------

__global__ __launch_bounds__(256)
void iso_k22_ainz(const float* __restrict__ maskT, const float* __restrict__ maskV,
                  const float* __restrict__ maskW,
                  const float* __restrict__ dyu,
                  const float* __restrict__ dzt, const float* __restrict__ dzw,
                  const float* __restrict__ zt,
                  const float* __restrict__ temp, const float* __restrict__ salt,
                  const float* __restrict__ Kiso,
                  float* __restrict__ K22, float* __restrict__ Ainz)
{
  __shared__ float smem[3 * NZ + NY];
  float* s_dzw = smem;
  float* s_dzt = smem + NZ;
  float* s_zt  = smem + 2 * NZ;
  float* s_dyu = smem + 3 * NZ;

  const int tid = threadIdx.x, nt = blockDim.x;
  const uint32_t sb = (uint32_t)(uintptr_t)smem;
  async_stage(sb,               dzw, NZ, tid, nt);
  async_stage(sb + 4u * NZ,     dzt, NZ, tid, nt);
  async_stage(sb + 8u * NZ,     zt,  NZ, tid, nt);
  async_stage(sb + 12u * NZ,    dyu, NY, tid, nt);
  async_wait_and_sync();

  int n = blockIdx.x * nt + tid;
  if (n >= NPT) return;
  int k = n % NZ; int r = n / NZ; int j = r % NY; int i = r / NY;
  size_t c = idx3(i, j, k);

  bool inside = (i >= 2) & (i <= NX - 3) & (j >= 1) & (j <= NY - 3);
  if (!inside) {
    nt_store(&K22[c], 0.0f);
    nt_store4(Ainz + c * 4u, 0.f, 0.f, 0.f, 0.f);
    return;
  }

  float mV = maskV[c];
  float dyuj = s_dyu[j];
  float Tc = ts(temp, i, j, k),  Tn = ts(temp, i, j + 1, k);
  float Sc = ts(salt, i, j, k),  Sn = ts(salt, i, j + 1, k);
  float dTdyv = mV * (Tn - Tc) / dyuj;
  float dSdyv = mV * (Sn - Sc) / dyuj;

  float diffloc = (k >= 1)
    ? 0.25f * (Kiso[idx3(i, j, k)] + Kiso[idx3(i, j, k - 1)]
             + Kiso[idx3(i, j + 1, k)] + Kiso[idx3(i, j + 1, k - 1)])
    : 0.5f * (Kiso[idx3(i, j, 0)] + Kiso[idx3(i, j + 1, 0)]);

  float sumz = 0.f;
  float ai[2][2] = {{0.f, 0.f}, {0.f, 0.f}};
  float abszk = fabsf(s_zt[k]);

  #pragma unroll
  for (int kr = 0; kr < 2; ++kr) {
    const int ki = (kr == 1) ? 0 : 1;
    if (k < ki) continue;
    const int kz = (kr == 0) ? k - 1 : k;
    float dzwz = s_dzw[kz];
    #pragma unroll
    for (int jp = 0; jp < 2; ++jp) {
      int jj = j + jp;
      float mT   = maskT[idx3(i, jj, k)];
      float drdT = mT * drhodT(ts(temp, i, jj, k), abszk);
      float drdS = mT * (BETA_Sc * RHO0);
      float dTdzv = 0.f, dSdzv = 0.f;
      if (kz < NZ - 1) {
        float mWv = maskW[idx3(i, jj, kz)];
        dTdzv = mWv * (ts(temp, i, jj, kz + 1) - ts(temp, i, jj, kz)) / dzwz;
        dSdzv = mWv * (ts(salt, i, jj, kz + 1) - ts(salt, i, jj, kz)) / dzwz;
      }
      float drodyn = drdT * dTdyv + drdS * dSdyv;
      float drodzn = drdT * dTdzv + drdS * dSdzv;
      float syn = -drodyn / (fminf(0.f, drodzn) - EPSc);
      float tap = dm_taper(syn);
      sumz += dzwz * mV * fmaxf(K_ISO_STEEP, diffloc * tap);
      ai[jp][kr] = tap * syn * mV;
    }
  }
  nt_store(&K22[c], sumz / (4.0f * s_dzt[k]));
  nt_store4(Ainz + c * 4u, ai[0][0], ai[0][1], ai[1][0], ai[1][1]);
}

// ---------------- kernel 3: K_33, Ai_bx, Ai_by (top faces) ----------------

__global__ __launch_bounds__(256)
void iso_k33_aib(const float* __restrict__ maskT, const float* __restrict__ maskU,
                 const float* __restrict__ maskV, const float* __restrict__ maskW,
                 const float* __restrict__ dxt, const float* __restrict__ dxu,
                 const float* __restrict__ dyt, const float* __restrict__ dyu,
                 const float* __restrict__ cost, const float* __restrict__ cosu,
                 const float* __restrict__ dzw, const float* __restrict__ zt,
                 const float* __restrict__ temp, const float* __restrict__ salt,
                 const float* __restrict__ Kiso,
                 float* __restrict__ K33, float* __restrict__ Aibx,
                 float* __restrict__ Aiby)
{
  __shared__ float smem[2 * NZ + 2 * NX + 4 * NY];
  float* s_dzw  = smem;                            // NZ
  float* s_zt   = smem + NZ;                       // NZ
  float* s_dxu  = smem + 2 * NZ;                   // NX
  float* s_dxt  = smem + 2 * NZ + NX;              // NX
  float* s_dyu  = smem + 2 * NZ + 2 * NX;          // NY
  float* s_dyt  = smem + 2 * NZ + 2 * NX + NY;     // NY
  float* s_cost = smem + 2 * NZ + 2 * NX + 2 * NY; // NY
  float* s_cosu = smem + 2 * NZ + 2 * NX + 3 * NY; // NY

  const int tid = threadIdx.x, nt = blockDim.x;
  const uint32_t sb = (uint32_t)(uintptr_t)smem;
  async_stage(sb,                               dzw,  NZ, tid, nt);
  async_stage(sb + 4u * NZ,                     zt,   NZ, tid, nt);
  async_stage(sb + 8u * NZ,                     dxu,  NX, tid, nt);
  async_stage(sb + 8u * NZ + 4u * NX,           dxt,  NX, tid, nt);
  async_stage(sb + 8u * NZ + 8u * NX,           dyu,  NY, tid, nt);
  async_stage(sb + 8u * NZ + 8u * NX + 4u * NY, dyt,  NY, tid, nt);
  async_stage(sb + 8u * NZ + 8u * NX + 8u * NY, cost, NY, tid, nt);
  async_stage(sb + 8u * NZ + 8u * NX + 12u * NY, cosu, NY, tid, nt);
  async_wait_and_sync();

  int n = blockIdx.x * nt + tid;
  if (n >= NPT) return;
  int k = n % NZ; int r = n / NZ; int j = r % NY; int i = r / NY;
  size_t c = idx3(i, j, k);

  bool inside = (i >= 2) & (i <= NX - 3) & (j >= 2) & (j <= NY - 3) & (k <= NZ - 2);
  if (!inside) {
    nt_store(&K33[c], 0.0f);
    nt_store4(Aibx + c * 4u, 0.f, 0.f, 0.f, 0.f);
    nt_store4(Aiby + c * 4u, 0.f, 0.f, 0.f, 0.f);
    return;
  }

  float mW   = maskW[c];
  float dzwk = s_dzw[k];
  float dTdzc = mW * (ts(temp, i, j, k + 1) - ts(temp, i, j, k)) / dzwk;
  float dSdzc = mW * (ts(salt, i, j, k + 1) - ts(salt, i, j, k)) / dzwk;
  float Kc    = Kiso[c];
  float costj = s_cost[j];

  float sumx = 0.f, sumy = 0.f;
  float abx[2][2], aby[2][2];

  #pragma unroll
  for (int kr = 0; kr < 2; ++kr) {
    int kzb = k + kr;
    float mT    = maskT[idx3(i, j, kzb)];
    float drdTb = mT * drhodT(ts(temp, i, j, kzb), fabsf(s_zt[kzb]));
    float drdSb = mT * (BETA_Sc * RHO0);
    float drodzb = drdTb * dTdzc + drdSb * dSdzc;
    float denom  = fminf(0.f, drodzb) - EPSc;

    #pragma unroll
    for (int ip = 0; ip < 2; ++ip) {
      int ii = i - 1 + ip;
      float mU  = maskU[idx3(ii, j, kzb)];
      float fxi = s_dxu[ii] * costj;
      float dTdxv = mU * (ts(temp, ii + 1, j, kzb) - ts(temp, ii, j, kzb)) / fxi;
      float dSdxv = mU * (ts(salt, ii + 1, j, kzb) - ts(salt, ii, j, kzb)) / fxi;
      float drodxb = drdTb * dTdxv + drdSb * dSdxv;
      float sxb = -drodxb / denom;
      float tap = dm_taper(sxb);
      sumx += s_dxu[ii] * Kc * tap * sxb * sxb * mW;
      abx[ip][kr] = tap * sxb * mW;
    }

    #pragma unroll
    for (int jp = 0; jp < 2; ++jp) {
      int jj = j - 1 + jp;
      float mV   = maskV[idx3(i, jj, kzb)];
      float dyuj = s_dyu[jj];
      float dTdyv = mV * (ts(temp, i, jj + 1, kzb) - ts(temp, i, jj, kzb)) / dyuj;
      float dSdyv = mV * (ts(salt, i, jj + 1, kzb) - ts(salt, i, jj, kzb)) / dyuj;
      float drodyb = drdTb * dTdyv + drdSb * dSdyv;
      float syb = -drodyb / denom;
      float tap = dm_taper(syb);
      sumy += s_cosu[jj] * dyuj * Kc * tap * syb * syb * mW;
      aby[jp][kr] = tap * syb * mW;
    }
  }

  nt_store(&K33[c], sumx / (4.0f * s_dxt[i]) + sumy / (4.0f * s_dyt[j] * costj));
  nt_store4(Aibx + c * 4u, abx[0][0], abx[0][1], abx[1][0], abx[1][1]);
  nt_store4(Aiby + c * 4u, aby[0][0], aby[0][1], aby[1][0], aby[1][1]);
}

// ---------------- launch ---------------------------------------------------

extern "C" void kernel_launch(void* const* d_in, const int* in_sizes, int n_in,
                              void* d_out, int out_size, void* d_ws, size_t ws_size,
                              hipStream_t stream) {
  const float* maskT = (const float*)d_in[0];
  const float* maskU = (const float*)d_in[1];
  const float* maskV = (const float*)d_in[2];
  const float* maskW = (const float*)d_in[3];
  const float* dxt   = (const float*)d_in[4];
  const float* dxu   = (const float*)d_in[5];
  const float* dyt   = (const float*)d_in[6];
  const float* dyu   = (const float*)d_in[7];
  const float* dzt   = (const float*)d_in[8];
  const float* dzw   = (const float*)d_in[9];
  const float* cost  = (const float*)d_in[10];
  const float* cosu  = (const float*)d_in[11];
  const float* salt  = (const float*)d_in[12];
  const float* temp  = (const float*)d_in[13];
  const float* zt    = (const float*)d_in[14];
  const float* Kiso  = (const float*)d_in[15];
  // d_in[16..22] are the zero-initialized output templates; unused.

  const size_t N = (size_t)NPT;
  float* K11  = (float*)d_out;
  float* K22  = K11 + N;
  float* K33  = K11 + 2 * N;
  float* Aiez = K11 + 3 * N;
  float* Ainz = K11 + 7 * N;
  float* Aibx = K11 + 11 * N;
  float* Aiby = K11 + 15 * N;

  dim3 blk(256);
  dim3 grd((NPT + 255) / 256);

  iso_k11_aiez<<<grd, blk, 0, stream>>>(maskT, maskU, maskW, dxu, cost,
                                        dzt, dzw, zt, temp, salt, Kiso,
                                        K11, Aiez);
  iso_k22_ainz<<<grd, blk, 0, stream>>>(maskT, maskV, maskW, dyu,
                                        dzt, dzw, zt, temp, salt, Kiso,
                                        K22, Ainz);
  iso_k33_aib<<<grd, blk, 0, stream>>>(maskT, maskU, maskV, maskW,
                                       dxt, dxu, dyt, dyu, cost, cosu,
                                       dzw, zt, temp, salt, Kiso,
                                       K33, Aibx, Aiby);
}